// CombinedLoss_38422777430448
// MI455X (gfx1250) — compile-verified
//
#include <hip/hip_runtime.h>

// CombinedLoss: 0.7*MSE + 0.3*softDTW(gamma=0.2), B=64, T=1024, C=8, fp32.
// One workgroup per batch. Blocked anti-diagonal DP over 16x16 tiles whose
// distance blocks are produced by V_WMMA_F32_16X16X4_F32 (K=8 via 2 chained
// WMMAs). Everything LDS-resident; 2 block barriers per tile-diagonal.

#define T_    1024
#define CCH   8
#define NB    64
#define NT    64            // 1024/16 tiles per side
#define NTHREADS 512
#define NWAVES   16
#define GAMMA_ 0.2f
#define ALPHA_ 0.7f
#define INF_   1000000000.0f

typedef float v2f __attribute__((ext_vector_type(2)));
typedef float v8f __attribute__((ext_vector_type(8)));

__device__ __forceinline__ float softmin3(float a, float b, float c) {
  // -g*log(sum exp(-x/g)) computed stably via min-shift, base-2 HW trans ops.
  const float k = 1.44269504088896f / GAMMA_;          // log2(e)/gamma
  const float g_ln2 = GAMMA_ * 0.693147180559945f;     // gamma*ln(2)
  float mn = fminf(a, fminf(b, c));
  float e = __builtin_amdgcn_exp2f((mn - a) * k)
          + __builtin_amdgcn_exp2f((mn - b) * k)
          + __builtin_amdgcn_exp2f((mn - c) * k);
  return mn - g_ln2 * __builtin_amdgcn_logf(e);        // v_log_f32 = log2
}

__global__ __launch_bounds__(NTHREADS) void sdtw_fused_kernel(
    const float* __restrict__ pred, const float* __restrict__ targ,
    float* __restrict__ ws) {
  const int b = blockIdx.x;
  const float* P = pred + (size_t)b * T_ * CCH;
  const float* Q = targ + (size_t)b * T_ * CCH;
  const int tid  = threadIdx.x;
  const int lane = tid & 31;
  const int wave = tid >> 5;
  const int half = lane >> 4;     // 0: lanes 0-15, 1: lanes 16-31
  const int n    = lane & 15;     // column within a 16x16 tile

  __shared__ float ROWs[T_];          // bottom row of last finished tile/col
  __shared__ float COLs[T_];          // right col of last finished tile/row
  __shared__ float CORN[NT];          // saved corners per tile-column
  __shared__ float X2[T_];
  __shared__ float Y2[T_];
  __shared__ float DT[NWAVES][2][256];// per-wave 16x16 distance tiles
  __shared__ float RED[NWAVES];

  // ---------- init: X2, Y2, fused per-batch SSE (for MSE) ----------
  float sse = 0.0f;
  for (int i = tid; i < T_; i += NTHREADS) {
    const float4* Pp = (const float4*)(P + i * CCH);
    const float4* Qp = (const float4*)(Q + i * CCH);
    float4 p0 = Pp[0], p1 = Pp[1], q0 = Qp[0], q1 = Qp[1];
    X2[i] = p0.x*p0.x + p0.y*p0.y + p0.z*p0.z + p0.w*p0.w +
            p1.x*p1.x + p1.y*p1.y + p1.z*p1.z + p1.w*p1.w;
    Y2[i] = q0.x*q0.x + q0.y*q0.y + q0.z*q0.z + q0.w*q0.w +
            q1.x*q1.x + q1.y*q1.y + q1.z*q1.z + q1.w*q1.w;
    float d0 = p0.x-q0.x, d1 = p0.y-q0.y, d2 = p0.z-q0.z, d3 = p0.w-q0.w;
    float d4 = p1.x-q1.x, d5 = p1.y-q1.y, d6 = p1.z-q1.z, d7 = p1.w-q1.w;
    sse += d0*d0+d1*d1+d2*d2+d3*d3+d4*d4+d5*d5+d6*d6+d7*d7;
  }
  for (int off = 16; off > 0; off >>= 1) sse += __shfl_down(sse, off, 32);
  if (lane == 0) RED[wave] = sse;
  __syncthreads();
  if (tid == 0) {
    float s = 0.0f;
    for (int w = 0; w < NWAVES; ++w) s += RED[w];
    ws[NB + b] = s;                 // per-batch sum of squared diffs
  }
  __syncthreads();

  // ---------- blocked anti-diagonal DP over 64x64 tile grid ----------
  for (int kk = 0; kk < 2 * NT - 1; ++kk) {
    const int Imin = (kk > NT - 1) ? (kk - (NT - 1)) : 0;
    const int Imax = (kk < NT - 1) ? kk : (NT - 1);
    const int ntiles = Imax - Imin + 1;

    // phase R: snapshot corners (CORN is rewritten by same-diagonal tiles)
    float cornSave[4];
    #pragma unroll
    for (int s = 0; s < 4; ++s) {
      const int idx = wave * 4 + s;
      cornSave[s] = INF_;
      if (idx < ntiles) {
        const int I = Imin + idx;
        const int J = kk - I;
        if (I == 0 && J == 0)      cornSave[s] = 0.0f;   // r[-1][-1] := 0
        else if (I > 0 && J > 0)   cornSave[s] = CORN[J - 1];
      }
    }
    __syncthreads();

    for (int pair = 0; pair < 2; ++pair) {
      const int idx0 = wave * 4 + pair * 2;

      // --- WMMA: distance tiles D = x2 + y2 - 2*A*B^T into LDS ---
      #pragma unroll
      for (int s = 0; s < 2; ++s) {
        const int idx = idx0 + s;              // wave-uniform condition
        if (idx < ntiles) {
          const int I = Imin + idx;
          const int J = kk - I;
          // A 16x4 f32: lane<16 rows 0-15 K={0,1}; lane>=16 K={2,3}
          const float* Ap = P + ((I << 4) + n) * CCH + half * 2;
          const float* Bp = Q + ((J << 4) + n) * CCH + half * 2;
          v2f a0 = *(const v2f*)(Ap);
          v2f b0 = *(const v2f*)(Bp);
          v2f a1 = *(const v2f*)(Ap + 4);
          v2f b1 = *(const v2f*)(Bp + 4);
          v8f acc = {};
          acc = __builtin_amdgcn_wmma_f32_16x16x4_f32(
              false, a0, false, b0, (short)0, acc, false, false);
          acc = __builtin_amdgcn_wmma_f32_16x16x4_f32(
              false, a1, false, b1, (short)0, acc, false, false);
          const float y2v = Y2[(J << 4) + n];
          #pragma unroll
          for (int r = 0; r < 8; ++r) {        // C/D: M = r + 8*half, N = n
            const int mrow = r + (half << 3);
            DT[wave][s][mrow * 16 + n] =
                X2[(I << 4) + mrow] + y2v - 2.0f * acc[r];
          }
        }
      }

      // --- dual-tile in-register wavefront DP (lanes 0-15 / 16-31) ---
      const int myIdx = idx0 + half;
      const bool tvalid = (myIdx < ntiles);
      const int I = Imin + (tvalid ? myIdx : 0);
      const int J = kk - I;
      const int t0 = I << 4;
      const int s0 = J << 4;
      const float topn   = (tvalid && I > 0) ? ROWs[s0 + n] : INF_;
      const float topnm1 = (tvalid && I > 0 && n > 0) ? ROWs[s0 + n - 1] : INF_;
      const float corner = cornSave[pair * 2 + half];
      const float oldRowLast = (tvalid && I > 0) ? ROWs[s0 + 15] : INF_;
      float rcur = INF_, rprev = INF_;
      #pragma unroll 1
      for (int p = 0; p < 31; ++p) {
        const float sl_cur  = __shfl_up(rcur, 1, 32);
        const float sl_prev = __shfl_up(rprev, 1, 32);
        const int m = p - n;                         // row this lane is on
        const bool act = tvalid && (m >= 0) && (m < 16);
        const int mc = (m < 0) ? 0 : ((m > 15) ? 15 : m);
        const float d = DT[wave][half][mc * 16 + n];
        const float lb  = (J > 0) ? COLs[t0 + mc] : INF_;               // left
        const float lbd = (J > 0) ? COLs[t0 + ((mc > 0) ? mc - 1 : 0)]  // diag
                                  : INF_;
        const float up   = (m == 0) ? topn : rcur;
        const float left = (n == 0) ? lb : sl_cur;
        const float diag = (m == 0) ? ((n == 0) ? corner : topnm1)
                                    : ((n == 0) ? lbd : sl_prev);
        const float rnew = d + softmin3(up, left, diag);
        if (act) {
          rprev = rcur;
          rcur  = rnew;
          if (n == 15) COLs[t0 + m] = rnew;   // publish right column
        }
      }
      if (tvalid) {
        ROWs[s0 + n] = rcur;                  // publish bottom row r[15][n]
        if (n == 15) CORN[J] = oldRowLast;    // corner for tile (I, J+1)
      }
    }
    __syncthreads();
  }

  if (tid == 0) ws[b] = ROWs[T_ - 1];         // r[1023][1023] for this batch
}

__global__ void combine_kernel(const float* __restrict__ ws,
                               float* __restrict__ out) {
  __shared__ float sd[NB];
  __shared__ float ssq[NB];
  const int t = threadIdx.x;
  sd[t]  = ws[t];
  ssq[t] = ws[NB + t];
  __syncthreads();
  if (t == 0) {
    float a = 0.0f, c = 0.0f;
    for (int i = 0; i < NB; ++i) { a += sd[i]; c += ssq[i]; }
    const float mse  = c / (float)(NB * T_ * CCH);
    const float sdtw = a / (float)NB;
    out[0] = ALPHA_ * mse + (1.0f - ALPHA_) * sdtw;
  }
}

extern "C" void kernel_launch(void* const* d_in, const int* in_sizes, int n_in,
                              void* d_out, int out_size, void* d_ws,
                              size_t ws_size, hipStream_t stream) {
  (void)in_sizes; (void)n_in; (void)out_size; (void)ws_size;
  const float* pred = (const float*)d_in[0];
  const float* targ = (const float*)d_in[1];
  float* out = (float*)d_out;
  float* ws  = (float*)d_ws;   // ws[0..63]=sdtw per batch, ws[64..127]=sse
  sdtw_fused_kernel<<<NB, NTHREADS, 0, stream>>>(pred, targ, ws);
  combine_kernel<<<1, NB, 0, stream>>>(ws, out);
}